// SparseGLUAoE_36464272343324
// MI455X (gfx1250) — compile-verified
//
#include <hip/hip_runtime.h>
#include <stdint.h>

// ---------------------------------------------------------------------------
// Grouped GLU FFN (AoE MoE, capacity-balanced), all-f32 on MI455X (gfx1250).
//   Phase A: h[e,t,f] = silu(acts.wup^T) * (x.v1^T)      -> d_ws (128 MB)
//   Phase B: out[e,t,h'] = h.w2                           (K=4096 reduction)
// Roofline: 155 GFLOP vs 360 MB traffic => AI ~430 FLOP/B: compute-bound at
// f32 matrix rates, so V_WMMA_F32_16X16X4_F32 is the limiting resource.
// Data movement now uses the CDNA5 Tensor Data Mover: one TENSOR_LOAD_TO_LDS
// per 64x64 tile, DMA direct to LDS with hardware padding (4 DWORDs per 64 =>
// bank-conflict-free 68-float rows), double-buffered and overlapped with the
// WMMA loop via TENSORcnt.  h (128 MB) stays L2-resident (192 MB) for B.
// ---------------------------------------------------------------------------

typedef float v2f __attribute__((ext_vector_type(2)));
typedef float v8f __attribute__((ext_vector_type(8)));
typedef unsigned int v4u __attribute__((ext_vector_type(4)));
typedef int v4i __attribute__((ext_vector_type(4)));
typedef int v8i __attribute__((ext_vector_type(8)));

#define E_   8
#define H_   1024
#define F_   4096
#define DL_  256
#define TPE_ 1024

#define TM   64            // token tile (rows)
#define TN   64            // output-col tile
#define KC   64            // K chunk staged through LDS
#define LDSP 68            // padded LDS row stride in floats (TDM pad: 4/64)

__device__ __forceinline__ v8f zero8() {
  v8f z;
#pragma unroll
  for (int i = 0; i < 8; ++i) z[i] = 0.0f;
  return z;
}

// Generic shared pointer -> LDS byte offset (LDS aperture: addr[31:0]).
__device__ __forceinline__ uint32_t lds_off(const void* p) {
  return (uint32_t)(uintptr_t)p;
}

// ---------------------------------------------------------------------------
// TDM: issue one async 2-D tile load, 64 rows x 64 f32 cols, from row-major
// global memory (row stride = rowStrideFloats elements) into LDS at
// ldsByteOff.  pad_enable inserts 4 DWORDs every 64 DWORDs so rows land at
// the LDSP=68 conflict-free stride.  Descriptor per ISA 8.3/8.4; groups 2/3
// zero (<=2-D tile), workgroup_mask=0 (not in a cluster).
// ---------------------------------------------------------------------------
__device__ __forceinline__ void tdm_load_tile64(uint32_t ldsByteOff,
                                                const float* gptr,
                                                int rowStrideFloats) {
  uint64_t ga = (uint64_t)(uintptr_t)gptr;
  v4u g0;
  g0[0] = 1u;                                    // count=1 (valid user desc)
  g0[1] = ldsByteOff;                            // lds_addr
  g0[2] = (uint32_t)(ga & 0xFFFFFFFFu);          // global_addr[31:0]
  g0[3] = (uint32_t)((ga >> 32) & 0x01FFFFFFu)   // global_addr[56:32]
        | (2u << 30);                            // type = 2 ("image")
  v8i g1;
  g1[0] = (2 << 16)                              // data_size = 4 bytes
        | (1 << 20)                              // pad_enable
        | (5 << 22)                              // pad_interval: every 64 DW
        | (3 << 25);                             // pad_amount: 4 DWORDs
  g1[1] = (KC & 0xFFFF) << 16;                   // tensor_dim0[15:0] @ [63:48]
  g1[2] = (64 & 0xFFFF) << 16;                   // tensor_dim1[15:0] @ [95:80]
  g1[3] = (KC << 16);                            // tile_dim0 = 64 @ [127:112]
  g1[4] = 64;                                    // tile_dim1 = 64; tile_dim2=0
  g1[5] = rowStrideFloats;                       // tensor_dim0_stride lo32
  g1[6] = 0;                                     // stride hi16 / dim1_stride
  g1[7] = 0;
  v4i zz = {0, 0, 0, 0};
#if defined(__clang_major__) && (__clang_major__ >= 23)
  v8i zz8 = {0, 0, 0, 0, 0, 0, 0, 0};
  __builtin_amdgcn_tensor_load_to_lds(g0, g1, zz, zz, zz8, 0);
#else
  __builtin_amdgcn_tensor_load_to_lds(g0, g1, zz, zz, 0);
#endif
}

// ---------------------------------------------------------------------------
// WMMA cores (unchanged from round 1 — these already lower to
// v_wmma_f32_16x16x4_f32 with ds_load_2addr_b64 fragment feeds).
// Fragment layouts per ISA 7.12.2: A lane=(M=lane&15, K=2*(lane>>4)+{0,1});
// B lane=(N=lane&15, K=2*(lane>>4)+{0,1}); C vgpr r = row r (+8 hi lanes).
// ---------------------------------------------------------------------------
__device__ __forceinline__ void mma_chunk_bt(const float* a_lds,
                                             const float* b_lds,
                                             int w, int lane, v8f acc[4]) {
  const int hi = lane >> 4, lo = lane & 15;
  const float* ap = a_lds + (16 * w + lo) * LDSP + 2 * hi;
#pragma unroll
  for (int k = 0; k < KC; k += 4) {
    v2f af = *(const v2f*)(ap + k);
#pragma unroll
    for (int c = 0; c < 4; ++c) {
      v2f bf = *(const v2f*)(b_lds + (16 * c + lo) * LDSP + k + 2 * hi);
      acc[c] = __builtin_amdgcn_wmma_f32_16x16x4_f32(
          false, af, false, bf, (short)0, acc[c], false, false);
    }
  }
}

__device__ __forceinline__ void mma_chunk_bn(const float* a_lds,
                                             const float* b_lds,
                                             int w, int lane, v8f acc[4]) {
  const int hi = lane >> 4, lo = lane & 15;
  const float* ap = a_lds + (16 * w + lo) * LDSP + 2 * hi;
#pragma unroll
  for (int k = 0; k < KC; k += 4) {
    v2f af = *(const v2f*)(ap + k);
#pragma unroll
    for (int c = 0; c < 4; ++c) {
      const float* bp = b_lds + (k + 2 * hi) * LDSP + 16 * c + lo;
      v2f bf;
      bf[0] = bp[0];
      bf[1] = bp[LDSP];
      acc[c] = __builtin_amdgcn_wmma_f32_16x16x4_f32(
          false, af, false, bf, (short)0, acc[c], false, false);
    }
  }
}

// ------------------------------- Phase A -----------------------------------
// grid = (F/64, TPE/64, E), block = 128 (4 waves; wave w owns 16x64 strip)
__global__ __launch_bounds__(128) void glu_phaseA(
    const float* __restrict__ x, const float* __restrict__ acts,
    const float* __restrict__ wup, const float* __restrict__ v1,
    float* __restrict__ hbuf) {
  __shared__ float a_lds[2][TM * LDSP];
  __shared__ float b_lds[2][TN * LDSP];

  const int tid = threadIdx.x;
  const int w = tid >> 5, lane = tid & 31;
  const int f0 = blockIdx.x * TN;
  const int t0 = blockIdx.y * TM;
  const int e  = blockIdx.z;

  const float* ag = acts + (size_t)(e * TPE_ + t0) * DL_;
  const float* xg = x    + (size_t)(e * TPE_ + t0) * H_;
  const float* wu = wup  + (size_t)e * F_ * DL_ + (size_t)f0 * DL_;
  const float* vv = v1   + (size_t)e * F_ * H_  + (size_t)f0 * H_;

  v8f acc1[4], acc2[4];
#pragma unroll
  for (int c = 0; c < 4; ++c) { acc1[c] = zero8(); acc2[c] = zero8(); }

  int buf = 0;

  // --- GEMM 1: x1 = acts . wup^T (K = DL = 256), TDM double-buffered ------
  if (tid < 32) {
    tdm_load_tile64(lds_off(a_lds[0]), ag, DL_);
    tdm_load_tile64(lds_off(b_lds[0]), wu, DL_);
  }
  for (int k0 = 0; k0 < DL_; k0 += KC) {
    if (tid < 32) {
      if (k0 + KC < DL_) {
        tdm_load_tile64(lds_off(a_lds[buf ^ 1]), ag + k0 + KC, DL_);
        tdm_load_tile64(lds_off(b_lds[buf ^ 1]), wu + k0 + KC, DL_);
        __builtin_amdgcn_s_wait_tensorcnt(2);   // chunk k0 has landed
      } else {
        __builtin_amdgcn_s_wait_tensorcnt(0);
      }
    }
    __syncthreads();                            // publish LDS tile to WG
    mma_chunk_bt(a_lds[buf], b_lds[buf], w, lane, acc1);
    __syncthreads();                            // protect buf from next DMA
    buf ^= 1;
  }

  // --- GEMM 2: x2 = x . v1^T (K = H = 1024), TDM double-buffered ----------
  if (tid < 32) {
    tdm_load_tile64(lds_off(a_lds[buf]), xg, H_);
    tdm_load_tile64(lds_off(b_lds[buf]), vv, H_);
  }
  for (int k0 = 0; k0 < H_; k0 += KC) {
    if (tid < 32) {
      if (k0 + KC < H_) {
        tdm_load_tile64(lds_off(a_lds[buf ^ 1]), xg + k0 + KC, H_);
        tdm_load_tile64(lds_off(b_lds[buf ^ 1]), vv + k0 + KC, H_);
        __builtin_amdgcn_s_wait_tensorcnt(2);
      } else {
        __builtin_amdgcn_s_wait_tensorcnt(0);
      }
    }
    __syncthreads();
    mma_chunk_bt(a_lds[buf], b_lds[buf], w, lane, acc2);
    __syncthreads();
    buf ^= 1;
  }

  // --- h = silu(x1) * x2 ; store to workspace -----------------------------
  const int hi = lane >> 4, lo = lane & 15;
#pragma unroll
  for (int c = 0; c < 4; ++c) {
#pragma unroll
    for (int r = 0; r < 8; ++r) {
      float a = acc1[c][r];
      float b = acc2[c][r];
      float hv = (a / (1.0f + __expf(-a))) * b;   // silu(a) * b
      int row = t0 + 16 * w + r + 8 * hi;
      int col = f0 + 16 * c + lo;
      hbuf[(size_t)(e * TPE_ + row) * F_ + col] = hv;
    }
  }
}

// ------------------------------- Phase B -----------------------------------
// grid = (H/64, TPE/64, E), block = 128 (4 waves)
__global__ __launch_bounds__(128) void glu_phaseB(
    const float* __restrict__ hbuf, const float* __restrict__ w2,
    float* __restrict__ out) {
  __shared__ float a_lds[2][TM * LDSP];
  __shared__ float b_lds[2][KC * LDSP];

  const int tid = threadIdx.x;
  const int w = tid >> 5, lane = tid & 31;
  const int h0 = blockIdx.x * TN;
  const int t0 = blockIdx.y * TM;
  const int e  = blockIdx.z;

  const float* ha = hbuf + (size_t)(e * TPE_ + t0) * F_;       // [t][k]
  const float* wb = w2 + (size_t)e * F_ * H_ + h0;             // [k][n]

  v8f acc[4];
#pragma unroll
  for (int c = 0; c < 4; ++c) acc[c] = zero8();

  int buf = 0;
  if (tid < 32) {
    tdm_load_tile64(lds_off(a_lds[0]), ha, F_);
    tdm_load_tile64(lds_off(b_lds[0]), wb, H_);
  }
  for (int k0 = 0; k0 < F_; k0 += KC) {
    if (tid < 32) {
      if (k0 + KC < F_) {
        tdm_load_tile64(lds_off(a_lds[buf ^ 1]), ha + k0 + KC, F_);
        tdm_load_tile64(lds_off(b_lds[buf ^ 1]),
                        wb + (size_t)(k0 + KC) * H_, H_);
        __builtin_amdgcn_s_wait_tensorcnt(2);
      } else {
        __builtin_amdgcn_s_wait_tensorcnt(0);
      }
    }
    __syncthreads();
    mma_chunk_bn(a_lds[buf], b_lds[buf], w, lane, acc);
    __syncthreads();
    buf ^= 1;
  }

  const int hi = lane >> 4, lo = lane & 15;
#pragma unroll
  for (int c = 0; c < 4; ++c) {
#pragma unroll
    for (int r = 0; r < 8; ++r) {
      int row = t0 + 16 * w + r + 8 * hi;
      int col = h0 + 16 * c + lo;
      out[(size_t)(e * TPE_ + row) * H_ + col] = acc[c][r];
    }
  }
}

// ---------------------------------------------------------------------------
extern "C" void kernel_launch(void* const* d_in, const int* in_sizes, int n_in,
                              void* d_out, int out_size, void* d_ws,
                              size_t ws_size, hipStream_t stream) {
  const float* x    = (const float*)d_in[0];
  const float* acts = (const float*)d_in[1];
  const float* wup  = (const float*)d_in[2];
  const float* v1   = (const float*)d_in[3];
  const float* w2   = (const float*)d_in[4];
  // d_in[5] = tokens_per_expert (==1024, folded into compile-time constants)

  float* hbuf = (float*)d_ws;  // requires E*TPE*F*4 = 128 MB of workspace
  float* out  = (float*)d_out;

  dim3 ga(F_ / TN, TPE_ / TM, E_);  // 64 x 16 x 8
  glu_phaseA<<<ga, 128, 0, stream>>>(x, acts, wup, v1, hbuf);

  dim3 gb(H_ / TN, TPE_ / TM, E_);  // 16 x 16 x 8
  glu_phaseB<<<gb, 128, 0, stream>>>(hbuf, w2, out);
}